// VtraceVLoss_33036888440950
// MI455X (gfx1250) — compile-verified
//
#include <hip/hip_runtime.h>

// ---------------------------------------------------------------------------
// V-trace value loss on MI455X (gfx1250, wave32).
//
// With CLIP_RHO == CLIP_C == CLIP_RHO_PG == 1:
//   rho   = min(1, exp(lp - olp))
//   a[i]  = terminals[i] * rho[i]
//   d[i]  = rho[i] * (rewards[i] + terminals[i]*next_values[i] - values[i])
//   y[i]  = a[i]*y[i+1] + d[i]      (backward scan, y[T] = 0)
//   pg_advantages[i] == y[i]        (algebraic identity, see analysis)
//   critic_loss      == mean(y^2)
//
// Memory-bound: 96 MB inputs (fits 192 MB L2) + 16 MB output.
// 3-level chunked affine scan, deterministic (no float atomics).
// ---------------------------------------------------------------------------

#define T_TOTAL       4194304
#define ITEMS         16
#define BLOCK_THREADS 256
#define CHUNK         (BLOCK_THREADS * ITEMS)   // 4096 elements per block
#define NBLOCKS       (T_TOTAL / CHUNK)         // 1024 blocks

typedef __attribute__((ext_vector_type(2))) float v2f;
typedef __attribute__((ext_vector_type(8))) float v8f;

// Compute a[i], d[i] for one thread's 16-element segment (vectorized loads).
__device__ __forceinline__ void compute_ad(
    const float* __restrict__ lp, const float* __restrict__ olp,
    const float* __restrict__ vv, const float* __restrict__ nv,
    const float* __restrict__ rw, const float* __restrict__ tm,
    int base, float a[ITEMS], float d[ITEMS])
{
#pragma unroll
    for (int j = 0; j < ITEMS / 4; ++j) {
        const int o = base + 4 * j;
        float4 L  = *reinterpret_cast<const float4*>(lp  + o);
        float4 O  = *reinterpret_cast<const float4*>(olp + o);
        float4 V  = *reinterpret_cast<const float4*>(vv  + o);
        float4 N  = *reinterpret_cast<const float4*>(nv  + o);
        float4 R  = *reinterpret_cast<const float4*>(rw  + o);
        float4 Tm = *reinterpret_cast<const float4*>(tm  + o);
#define VT_ELEM(k, Lc, Oc, Vc, Nc, Rc, Tc)              \
        {                                               \
            float ratio = __expf((Lc) - (Oc));          \
            float rho   = fminf(1.0f, ratio);           \
            float td    = (Rc) + (Tc) * (Nc);           \
            d[4*j + k]  = rho * (td - (Vc));            \
            a[4*j + k]  = (Tc) * rho;                   \
        }
        VT_ELEM(0, L.x, O.x, V.x, N.x, R.x, Tm.x)
        VT_ELEM(1, L.y, O.y, V.y, N.y, R.y, Tm.y)
        VT_ELEM(2, L.z, O.z, V.z, N.z, R.z, Tm.z)
        VT_ELEM(3, L.w, O.w, V.w, N.w, R.w, Tm.w)
#undef VT_ELEM
    }
}

// Pass 1: per-block affine composition (A_blk, B_blk): y_blockStart = A*carry + B.
__global__ void __launch_bounds__(BLOCK_THREADS)
vtrace_pass1(const float* __restrict__ lp, const float* __restrict__ olp,
             const float* __restrict__ vv, const float* __restrict__ nv,
             const float* __restrict__ rw, const float* __restrict__ tm,
             float2* __restrict__ blkAB)
{
    const int b = blockIdx.x, t = threadIdx.x;
    const int lane = t & 31, w = t >> 5;
    const int base = b * CHUNK + t * ITEMS;

    float a[ITEMS], d[ITEMS];
    compute_ad(lp, olp, vv, nv, rw, tm, base, a, d);

    // Local segment composition (apply from high index down).
    float A = 1.0f, B = 0.0f;
#pragma unroll
    for (int i = ITEMS - 1; i >= 0; --i) { B = fmaf(a[i], B, d[i]); A *= a[i]; }

    // Wave32 inclusive *suffix* scan of affine maps: S_t = M_t ∘ S_{t+o}.
#pragma unroll
    for (int o = 1; o < 32; o <<= 1) {
        float Ao = __shfl_down(A, (unsigned)o, 32);
        float Bo = __shfl_down(B, (unsigned)o, 32);
        if (lane + o < 32) { B = fmaf(A, Bo, B); A *= Ao; }
    }

    __shared__ float wA[BLOCK_THREADS / 32], wB[BLOCK_THREADS / 32];
    if (lane == 0) { wA[w] = A; wB[w] = B; }
    __syncthreads();

    if (t == 0) {
        float CA = 1.0f, CB = 0.0f;
        for (int q = (BLOCK_THREADS / 32) - 1; q >= 0; --q) {
            CB = fmaf(wA[q], CB, wB[q]);
            CA *= wA[q];
        }
        blkAB[b] = make_float2(CA, CB);
    }
}

// Pass 2: suffix-scan the NBLOCKS block compositions; carry[b] = y at end of block b.
__global__ void __launch_bounds__(NBLOCKS)
vtrace_pass2(const float2* __restrict__ blkAB, float* __restrict__ carry)
{
    __shared__ float sA[2][NBLOCKS], sB[2][NBLOCKS];
    const int t = threadIdx.x;
    float2 ab = blkAB[t];
    sA[0][t] = ab.x; sB[0][t] = ab.y;
    __syncthreads();

    int cur = 0;
    for (int o = 1; o < NBLOCKS; o <<= 1) {
        int nxt = cur ^ 1;
        float A = sA[cur][t], B = sB[cur][t];
        if (t + o < NBLOCKS) {
            float A2 = sA[cur][t + o], B2 = sB[cur][t + o];
            B = fmaf(A, B2, B);   // (A,B) ∘ (A2,B2)
            A = A * A2;
        }
        sA[nxt][t] = A; sB[nxt][t] = B;
        __syncthreads();
        cur = nxt;
    }
    // carry[b] = y at blockEnd = inclusive suffix of block b+1, evaluated at 0.
    carry[t] = (t + 1 < NBLOCKS) ? sB[cur][t + 1] : 0.0f;
}

// Pass 3: replay with correct carries; write y (== pg_advantages), per-block sum(y^2).
__global__ void __launch_bounds__(BLOCK_THREADS)
vtrace_pass3(const float* __restrict__ lp, const float* __restrict__ olp,
             const float* __restrict__ vv, const float* __restrict__ nv,
             const float* __restrict__ rw, const float* __restrict__ tm,
             const float* __restrict__ carry, float* __restrict__ out,
             float* __restrict__ blkSumSq)
{
    const int b = blockIdx.x, t = threadIdx.x;
    const int lane = t & 31, w = t >> 5;
    const int base = b * CHUNK + t * ITEMS;

    float a[ITEMS], d[ITEMS];
    compute_ad(lp, olp, vv, nv, rw, tm, base, a, d);   // inputs now L2-resident

    float A = 1.0f, B = 0.0f;
#pragma unroll
    for (int i = ITEMS - 1; i >= 0; --i) { B = fmaf(a[i], B, d[i]); A *= a[i]; }

    // Wave32 inclusive suffix scan.
#pragma unroll
    for (int o = 1; o < 32; o <<= 1) {
        float Ao = __shfl_down(A, (unsigned)o, 32);
        float Bo = __shfl_down(B, (unsigned)o, 32);
        if (lane + o < 32) { B = fmaf(A, Bo, B); A *= Ao; }
    }
    // Exclusive suffix = neighbor's inclusive suffix (identity at lane 31).
    float Ax = __shfl_down(A, 1u, 32);
    float Bx = __shfl_down(B, 1u, 32);
    if (lane == 31) { Ax = 1.0f; Bx = 0.0f; }

    __shared__ float wA[BLOCK_THREADS / 32], wB[BLOCK_THREADS / 32];
    __shared__ float sS[BLOCK_THREADS / 32];
    if (lane == 0) { wA[w] = A; wB[w] = B; }    // wave totals (lane 0 holds S_0)
    __syncthreads();

    // Carry entering this wave's region: waves w+1..7 applied to block carry.
    float cw = carry[b];
    for (int q = (BLOCK_THREADS / 32) - 1; q > w; --q) cw = fmaf(wA[q], cw, wB[q]);
    // Carry entering this thread's segment.
    float v = fmaf(Ax, cw, Bx);

    float ss = 0.0f;
    float* yout = out + 1 + base;   // d_out[0] = loss, d_out[1..T] = pg_advantages
#pragma unroll
    for (int i = ITEMS - 1; i >= 0; --i) {
        v = fmaf(a[i], v, d[i]);                 // v = y[base + i]
        __builtin_nontemporal_store(v, yout + i); // write-once output, bypass cache
        ss = fmaf(v, v, ss);
    }

    // Block reduction of sum(y^2).
#pragma unroll
    for (int o = 16; o > 0; o >>= 1) ss += __shfl_down(ss, (unsigned)o, 32);
    if (lane == 0) sS[w] = ss;
    __syncthreads();
    if (t == 0) {
        float tot = 0.0f;
        for (int q = 0; q < BLOCK_THREADS / 32; ++q) tot += sS[q];
        blkSumSq[b] = tot;
    }
}

// Pass 4: reduce NBLOCKS partial sums -> critic_loss. Final 32-way reduction is
// done with V_WMMA_F32_16X16X4_F32 (A = partials as 16x4, B = ones 4x16):
// D[m][n] = q_m + q_{m+16}, so column 0 of D holds pairwise sums of all lanes.
__global__ void __launch_bounds__(256)
vtrace_pass4(const float* __restrict__ blkSumSq, float* __restrict__ out)
{
    __shared__ float s[256];
    const int t = threadIdx.x;
    const int lane = t & 31;

    float p = 0.0f;
    for (int j = t; j < NBLOCKS; j += 256) p += blkSumSq[j];
    s[t] = p;
    __syncthreads();

    // Every wave executes identically (EXEC must be all-1s for WMMA).
    float q = 0.0f;
#pragma unroll
    for (int j = 0; j < 8; ++j) q += s[lane + j * 32];

    v2f amat = {q, 0.0f};       // A 16x4: lane m -> A[m][0] (lanes 16-31 -> A[m][2])
    v2f bmat = {1.0f, 1.0f};    // B 4x16: all ones (layout-agnostic)
    v8f cmat = {};
    v8f dm = __builtin_amdgcn_wmma_f32_16x16x4_f32(
        /*neg_a=*/false, amat, /*neg_b=*/false, bmat,
        /*c_mod=*/(short)0, cmat, /*reuse_a=*/false, /*reuse_b=*/false);

    // Lane 0 regs: D[0..7][0]; lane 16 regs: D[8..15][0].
    float r = dm[0] + dm[1] + dm[2] + dm[3] + dm[4] + dm[5] + dm[6] + dm[7];
    float r16 = __shfl(r, 16, 32);
    if (t == 0) out[0] = (r + r16) * (1.0f / (float)T_TOTAL);
}

extern "C" void kernel_launch(void* const* d_in, const int* in_sizes, int n_in,
                              void* d_out, int out_size, void* d_ws, size_t ws_size,
                              hipStream_t stream)
{
    const float* lp  = (const float*)d_in[0];   // logpacs
    const float* olp = (const float*)d_in[1];   // old_logpacs
    const float* vv  = (const float*)d_in[2];   // values
    const float* nv  = (const float*)d_in[3];   // next_values
    const float* rw  = (const float*)d_in[4];   // rewards
    const float* tm  = (const float*)d_in[5];   // terminals (already * gamma)
    float* out = (float*)d_out;                 // [0]=critic_loss, [1..T]=pg_adv

    float2* blkAB   = (float2*)d_ws;                                        // 8 KB
    float*  carry   = (float*)((char*)d_ws + NBLOCKS * sizeof(float2));     // 4 KB
    float*  sumsq   = (float*)((char*)d_ws + NBLOCKS * (sizeof(float2) + sizeof(float)));

    vtrace_pass1<<<NBLOCKS, BLOCK_THREADS, 0, stream>>>(lp, olp, vv, nv, rw, tm, blkAB);
    vtrace_pass2<<<1, NBLOCKS, 0, stream>>>(blkAB, carry);
    vtrace_pass3<<<NBLOCKS, BLOCK_THREADS, 0, stream>>>(lp, olp, vv, nv, rw, tm,
                                                        carry, out, sumsq);
    vtrace_pass4<<<1, 256, 0, stream>>>(sumsq, out);
}